// BlockNet_64682207478362
// MI455X (gfx1250) — compile-verified
//
#include <hip/hip_runtime.h>
#include <math.h>

// ---- problem constants (match reference setup_inputs) ----
#define NN   5000      // nodes
#define FF   512       // features
#define EE   4000      // graph edges
#define E2C  2000      // L-block size
#define TT   1000000   // query edges
#define HH   16        // hidden

typedef float v2f __attribute__((ext_vector_type(2)));
typedef float v8f __attribute__((ext_vector_type(8)));

// ---------------- utility ----------------
__global__ void k_fill(float* p, float v, int n) {
    int i = blockIdx.x * blockDim.x + threadIdx.x;
    if (i < n) p[i] = v;
}

__global__ void k_relu(float* p, int n) {
    int i = blockIdx.x * blockDim.x + threadIdx.x;
    if (i < n) p[i] = fmaxf(p[i], 0.0f);
}

// deg = 1 + indegree
__global__ void k_deg_acc(float* deg, const int* ei, int E) {
    int e = blockIdx.x * blockDim.x + threadIdx.x;
    if (e < E) atomicAdd(&deg[ei[E + e]], 1.0f);
}

// ---------------- fp32 WMMA GEMM: C[M x 16] = A[M x K] @ B[K x 16] ----------------
// One wave per 16-row tile, V_WMMA_F32_16X16X4_F32 (full-precision fp32 matrix op).
// A fragment: lane holds A[row][k0 + 2*(lane/16) + {0,1}]  -> one b64 load
// B fragment: lane holds B[k0 + 2*(lane/16) + {0,1}][lane%16]
// C/D: vgpr v <-> row = row0 + v + 8*(lane/16), col = lane%16
//
// Tail handling: A row index is CLAMPED (not predicated) so all loads are
// unconditional (no EXEC save/restore in the inner loop); D rows >= M are
// masked at the store and each D row depends only on its own A row.
//
// Addressing is pointer-bump form (constant strides) so loads lower to
// immediate-offset global_load in s_clause groups; 2-stage double buffer
// hides load latency behind the WMMAs.  Requires K % 8 == 0 (K = 512, 16).
__global__ void k_gemm16(const float* __restrict__ A, const float* __restrict__ B,
                         float* __restrict__ C, int M, int K) {
    const int lane = threadIdx.x & 31;
    const int wave = threadIdx.x >> 5;
    const int tile = blockIdx.x * (blockDim.x >> 5) + wave;
    const int ntiles = (M + 15) >> 4;
    if (tile >= ntiles) return;               // wave-uniform: EXEC all-ones for WMMA

    const int m  = lane & 15;
    const int kh = lane >> 4;                 // which K-half this lane holds
    const int row0 = tile << 4;
    int ra = row0 + m;
    if (ra > M - 1) ra = M - 1;               // clamp (see note above)

    const v2f*   aq = (const v2f*)(A + (size_t)ra * K) + kh;   // +2 per k-step (K even -> 8B aligned)
    const float* bq = B + (size_t)(2 * kh) * 16 + m;           // +64 per k-step

    v8f acc = {};
    // 2-stage software pipeline: fragments for k0 and k0+4 in flight
    v2f a0 = aq[0];
    v2f b0; b0[0] = bq[0];   b0[1] = bq[16];
    v2f a1 = aq[2];
    v2f b1; b1[0] = bq[64];  b1[1] = bq[80];
    for (int k0 = 8; k0 < K; k0 += 8) {
        v2f a2 = aq[4];
        v2f b2; b2[0] = bq[128]; b2[1] = bq[144];
        v2f a3 = aq[6];
        v2f b3; b3[0] = bq[192]; b3[1] = bq[208];
        acc = __builtin_amdgcn_wmma_f32_16x16x4_f32(false, a0, false, b0,
                                                    (short)0, acc, false, false);
        acc = __builtin_amdgcn_wmma_f32_16x16x4_f32(false, a1, false, b1,
                                                    (short)0, acc, false, false);
        a0 = a2; b0 = b2; a1 = a3; b1 = b3;
        aq += 4; bq += 128;
    }
    acc = __builtin_amdgcn_wmma_f32_16x16x4_f32(false, a0, false, b0,
                                                (short)0, acc, false, false);
    acc = __builtin_amdgcn_wmma_f32_16x16x4_f32(false, a1, false, b1,
                                                (short)0, acc, false, false);

#pragma unroll
    for (int v = 0; v < 8; ++v) {
        const int r = row0 + v + 8 * kh;
        if (r < M) C[(size_t)r * 16 + m] = acc[v];
    }
}

// ---------------- GCN aggregation ----------------
// out[v][c] = h[v][c] * dinv[v]^2 + b[c]
__global__ void k_gcn_self(const float* __restrict__ h, const float* __restrict__ deg,
                           const float* __restrict__ b, float* __restrict__ out, int n) {
    int idx = blockIdx.x * blockDim.x + threadIdx.x;
    if (idx < n * HH) {
        int v = idx >> 4, c = idx & 15;
        float di = rsqrtf(deg[v]);
        out[idx] = h[idx] * di * di + b[c];
    }
}

// out[dst][c] += h[src][c] * dinv[src]*dinv[dst]
__global__ void k_gcn_edge(const float* __restrict__ h, const float* __restrict__ deg,
                           const int* __restrict__ ei, float* __restrict__ out, int E) {
    int idx = blockIdx.x * blockDim.x + threadIdx.x;
    if (idx < E * HH) {
        int e = idx >> 4, c = idx & 15;
        int s = ei[e], d = ei[E + e];
        float coef = rsqrtf(deg[s]) * rsqrtf(deg[d]);
        atomicAdd(&out[(size_t)d * HH + c], h[(size_t)s * HH + c] * coef);
    }
}

// row-renorm of g (rows of width HH) to max L2 norm 1
__global__ void k_grenorm(float* __restrict__ g, int n) {
    int v = blockIdx.x * blockDim.x + threadIdx.x;
    if (v >= n) return;
    float* row = g + (size_t)v * HH;
    float ss = 0.0f;
#pragma unroll
    for (int c = 0; c < HH; ++c) ss += row[c] * row[c];
    float nr = sqrtf(ss);
    float sc = (nr > 1.0f) ? (1.0f / (nr + 1e-7f)) : 1.0f;
#pragma unroll
    for (int c = 0; c < HH; ++c) row[c] *= sc;
}

// ---------------- mat-vec: y[row] = sum_c A[row*C + c] * v[c], one block per row ----------------
__global__ void k_rowdot(const float* __restrict__ A, const float* __restrict__ v,
                         float* __restrict__ y, int C) {
    __shared__ float sd[256];
    const int row = blockIdx.x;
    const int tid = threadIdx.x;
    float acc = 0.0f;
    for (int c = tid; c < C; c += 256) acc += A[(size_t)row * C + c] * v[c];
    sd[tid] = acc;
    __syncthreads();
    for (int off = 128; off > 0; off >>= 1) {
        if (tid < off) sd[tid] += sd[tid + off];
        __syncthreads();
    }
    if (tid == 0) y[row] = sd[0];
}

// ---------------- softmax-weighted row sum of the (implicit, rank-1 off-diagonal) sim ----------------
// sim row r (length 2*E2C):
//   r < E2C  : [ L0[r,:],  v0[r]*u1[:] ]
//   r >= E2C : [ v0[:]*u1[r-E2C],  L1[r-E2C,:] ]
// t[r] = sum_c exp(relu(s_rc)) * wsim[c] / sum_c exp(relu(s_rc))
__global__ void k_simrow(const float* __restrict__ L0, const float* __restrict__ L1,
                         const float* __restrict__ v0, const float* __restrict__ u1,
                         const float* __restrict__ wsim, float* __restrict__ t) {
    __shared__ float snum[256];
    __shared__ float sden[256];
    const int r = blockIdx.x;
    const int tid = threadIdx.x;
    float num = 0.0f, den = 0.0f;
    if (r < E2C) {
        const float vr = v0[r];
        for (int c = tid; c < 2 * E2C; c += 256) {
            float sc = (c < E2C) ? L0[(size_t)r * E2C + c] : vr * u1[c - E2C];
            float z = __expf(fmaxf(sc, 0.0f));
            num += z * wsim[c];
            den += z;
        }
    } else {
        const int rp = r - E2C;
        const float ur = u1[rp];
        for (int c = tid; c < 2 * E2C; c += 256) {
            float sc = (c < E2C) ? v0[c] * ur : L1[(size_t)rp * E2C + (c - E2C)];
            float z = __expf(fmaxf(sc, 0.0f));
            num += z * wsim[c];
            den += z;
        }
    }
    snum[tid] = num; sden[tid] = den;
    __syncthreads();
    for (int off = 128; off > 0; off >>= 1) {
        if (tid < off) { snum[tid] += snum[tid + off]; sden[tid] += sden[tid + off]; }
        __syncthreads();
    }
    if (tid == 0) t[r] = snum[0] / sden[0];
}

// s[n] = renorm_scalar( sum_f x[n][f] * ev[f] ), one block per row
__global__ void k_srow(const float* __restrict__ x, const float* __restrict__ ev,
                       float* __restrict__ s) {
    __shared__ float sd[256];
    const int n = blockIdx.x;
    const int tid = threadIdx.x;
    float acc = 0.0f;
    for (int f = tid; f < FF; f += 256) acc += x[(size_t)n * FF + f] * ev[f];
    sd[tid] = acc;
    __syncthreads();
    for (int off = 128; off > 0; off >>= 1) {
        if (tid < off) sd[tid] += sd[tid + off];
        __syncthreads();
    }
    if (tid == 0) {
        float v = sd[0];
        float a = fabsf(v);                 // row norm for DIM=1
        s[n] = (a > 1.0f) ? v / (a + 1e-7f) : v;
    }
}

// ---------------- T=1e6 edge MLP (memory-bound; weights staged in LDS) ----------------
__global__ void k_edge_mlp(const float* __restrict__ g, const float* __restrict__ s,
                           const int* __restrict__ te,
                           const float* __restrict__ w1, const float* __restrict__ b1,
                           const float* __restrict__ w2, const float* __restrict__ b2,
                           float* __restrict__ out, int T) {
    __shared__ float sw1[17 * 16];
    __shared__ float sb1[16];
    __shared__ float sw2[16];
    __shared__ float sb2;
    const int tid = threadIdx.x;
    for (int idx = tid; idx < 17 * 16 + 33; idx += blockDim.x) {
        if (idx < 272)       sw1[idx] = w1[idx];
        else if (idx < 288)  sb1[idx - 272] = b1[idx - 272];
        else if (idx < 304)  sw2[idx - 288] = w2[idx - 288];
        else                 sb2 = b2[0];
    }
    __syncthreads();

    const int t = blockIdx.x * blockDim.x + tid;
    if (t >= T) return;

    // prefetch the index stream ahead (global_prefetch_b8)
    if (t + 8192 < T) __builtin_prefetch(&te[2 * (t + 8192)], 0, 1);

    const int i = te[2 * t];
    const int j = te[2 * t + 1];

    float gi[16], gj[16];
    const float4* gip = reinterpret_cast<const float4*>(g + (size_t)i * HH);
    const float4* gjp = reinterpret_cast<const float4*>(g + (size_t)j * HH);
#pragma unroll
    for (int q = 0; q < 4; ++q) {
        *reinterpret_cast<float4*>(&gi[4 * q]) = gip[q];
        *reinterpret_cast<float4*>(&gj[4 * q]) = gjp[q];
    }

    float ds = s[i] - s[j];
    float f16 = 100.0f * ds * ds;           // feature column 16 (the d_sim column)

    float hid[16];
#pragma unroll
    for (int c = 0; c < 16; ++c) hid[c] = sb1[c] + f16 * sw1[16 * 16 + c];
#pragma unroll
    for (int k = 0; k < 16; ++k) {
        float d = gi[k] - gj[k];
        float fk = 0.001f * d * d;          // feature column k (the g_sq columns)
#pragma unroll
        for (int c = 0; c < 16; ++c) hid[c] += fk * sw1[k * 16 + c];
    }

    float acc = sb2;
#pragma unroll
    for (int c = 0; c < 16; ++c) {
        float hv = hid[c];
        hv = (hv > 0.0f) ? hv : 0.8f * hv;  // leaky_relu slope 0.8
        acc += hv * sw2[c];
    }
    float y = fminf(fabsf(acc), 40.0f);     // clip(|.|, 0, 40)
    out[t] = 1.0f / (__expf(y - 2.0f) + 1.0f);
}

// ---------------- driver ----------------
extern "C" void kernel_launch(void* const* d_in, const int* in_sizes, int n_in,
                              void* d_out, int out_size, void* d_ws, size_t ws_size,
                              hipStream_t stream) {
    const float* x     = (const float*)d_in[0];
    const float* L0    = (const float*)d_in[1];
    const float* L1    = (const float*)d_in[2];
    const float* w1g   = (const float*)d_in[3];   // conv1_w (512x16)
    const float* b1g   = (const float*)d_in[4];
    const float* w2g   = (const float*)d_in[5];   // conv2_w (16x16)
    const float* b2g   = (const float*)d_in[6];
    const float* wsim  = (const float*)d_in[7];   // weights_sim (4000x1)
    const float* Esim  = (const float*)d_in[8];   // embeddings_sim (512x4000)
    const float* Wod   = (const float*)d_in[9];   // (2000x2000)
    const float* wL0   = (const float*)d_in[10];  // (2000x1)
    const float* wL1   = (const float*)d_in[11];  // (2000x1)
    const float* lin1w = (const float*)d_in[12];  // (17x16)
    const float* lin1b = (const float*)d_in[13];
    const float* linw  = (const float*)d_in[14];  // (16x1)
    const float* linb  = (const float*)d_in[15];
    const int*   ei    = (const int*)d_in[16];    // edge_index (2xE)
    const int*   te    = (const int*)d_in[17];    // total_edges (Tx2)
    float* out = (float*)d_out;

    float* ws = (float*)d_ws;                     // element offsets (all 1024-float aligned)
    float* DEG = ws + 0;                          // 5000
    float* TMP = ws + 8  * 1024;                  // 80000  (gemm output, both layers)
    float* Hb  = ws + 96 * 1024;                  // 80000  (layer-1 activations)
    float* Gb  = ws + 184 * 1024;                 // 80000  (layer-2 activations -> renormed g)
    float* U0  = ws + 272 * 1024;                 // 2000
    float* U1  = ws + 276 * 1024;                 // 2000
    float* V0  = ws + 280 * 1024;                 // 2000
    float* Tt  = ws + 284 * 1024;                 // 4000
    float* EV  = ws + 292 * 1024;                 // 512
    float* Sv  = ws + 296 * 1024;                 // 5000

    const int B = 256;
    const int gemmBlocks = ((NN + 15) / 16 + 7) / 8;  // 8 waves/block, 1 tile/wave

    // degree (shared by both GCN layers)
    k_fill<<<(NN + B - 1) / B, B, 0, stream>>>(DEG, 1.0f, NN);
    k_deg_acc<<<(EE + B - 1) / B, B, 0, stream>>>(DEG, ei, EE);

    // GCN layer 1: h = relu(agg(x @ W1) + b1)
    k_gemm16<<<gemmBlocks, B, 0, stream>>>(x, w1g, TMP, NN, FF);
    k_gcn_self<<<(NN * HH + B - 1) / B, B, 0, stream>>>(TMP, DEG, b1g, Hb, NN);
    k_gcn_edge<<<(EE * HH + B - 1) / B, B, 0, stream>>>(TMP, DEG, ei, Hb, EE);
    k_relu<<<(NN * HH + B - 1) / B, B, 0, stream>>>(Hb, NN * HH);

    // GCN layer 2: g_emb = relu(agg(h @ W2) + b2), then row-renorm -> g
    k_gemm16<<<gemmBlocks, B, 0, stream>>>(Hb, w2g, TMP, NN, HH);
    k_gcn_self<<<(NN * HH + B - 1) / B, B, 0, stream>>>(TMP, DEG, b2g, Gb, NN);
    k_gcn_edge<<<(EE * HH + B - 1) / B, B, 0, stream>>>(TMP, DEG, ei, Gb, EE);
    k_relu<<<(NN * HH + B - 1) / B, B, 0, stream>>>(Gb, NN * HH);
    k_grenorm<<<(NN + B - 1) / B, B, 0, stream>>>(Gb, NN);

    // rank-1 factorization of rel_: u0 = L0@wL0, u1 = L1@wL1, v0 = Wod@u0
    k_rowdot<<<E2C, B, 0, stream>>>(L0, wL0, U0, E2C);
    k_rowdot<<<E2C, B, 0, stream>>>(L1, wL1, U1, E2C);
    k_rowdot<<<E2C, B, 0, stream>>>(Wod, U0, V0, E2C);

    // t = softmax(relu(sim), axis=1) @ weights_sim   (sim never materialized)
    k_simrow<<<2 * E2C, B, 0, stream>>>(L0, L1, V0, U1, wsim, Tt);

    // e = embeddings_sim @ t ; s = renorm(x @ e)
    k_rowdot<<<FF, B, 0, stream>>>(Esim, Tt, EV, 2 * E2C);
    k_srow<<<NN, B, 0, stream>>>(x, EV, Sv);

    // per query-edge MLP
    k_edge_mlp<<<(TT + B - 1) / B, B, 0, stream>>>(Gb, Sv, te, lin1w, lin1b, linw, linb,
                                                   out, TT);
}